// ObjectCondensationLoss_12678743458120
// MI455X (gfx1250) — compile-verified
//
#include <hip/hip_runtime.h>
#include <math.h>

#define BEV   8
#define NPT   262144
#define DIM   16
#define KSEG  256
#define MMAX  1024

#define ATTR_W         1.0f
#define REP_W          1.0f
#define BETA_POS_W     10.0f
#define BETA_NEG_SIG_W 3.0f
#define BETA_NEG_BG_W  6.0f
#define MARGIN_W       10.0f
#define THR            0.5f
#define MARGIN_C       0.3f
#define ALPHA_C        0.75f

// ---- workspace layout, 32-bit words per event ----
#define OFF_SUMF  0                       // float[256]  sum focal per instance
#define OFF_CNT   256                     // float[256]  cp count per instance
#define OFF_SIZE  512                     // float[256]  instance size
#define OFF_FIRST 768                     // int[256]    first cp index (init NPT)
#define OFF_SCAL  1024                    // float[16]   scalars, see below
#define OFF_CPEMB 1040                    // float[1024*16] CP embeddings (padded 0)
#define OFF_N2    (OFF_CPEMB + MMAX*DIM)  // float[1024] row norms
#define OFF_SELV  (OFF_N2 + MMAX)         // float[1024] select flags
#define EV_WORDS  (OFF_SELV + MMAX)       // 19472 words per event
// scalars: 0 n_valid, 1 n_cp, 2 n_ncp, 3 sum_ce0_ncp, 4 sum_posm, 5 sum_negm,
//          6 n_bg, 7 sum_ce0_bg, 8 attr, 9 rep_raw, 10 mv, 11 loss, 12 ok, 13 cp_count(int)

typedef float v2f __attribute__((ext_vector_type(2)));
typedef float v8f __attribute__((ext_vector_type(8)));

__global__ void k_init(float* __restrict__ ws) {
    int w = blockIdx.x * 256 + threadIdx.x;
    if (w >= BEV * EV_WORDS) return;
    int r = w % EV_WORDS;
    int* wi = (int*)ws;
    wi[w] = (r >= OFF_FIRST && r < OFF_FIRST + KSEG) ? NPT : 0;
}

// ---- per-point streaming stats: BCE terms + per-instance segment sums ----
__global__ __launch_bounds__(256) void k_stats(const float* __restrict__ beta,
                                               const int* __restrict__ sid,
                                               const unsigned char* __restrict__ cp,
                                               float* __restrict__ ws) {
    __shared__ float s_sumf[KSEG], s_cnt[KSEG], s_size[KSEG];
    __shared__ int   s_first[KSEG];
    __shared__ float s_scal[8];
    int tid = threadIdx.x;
    int e = blockIdx.y;
    s_sumf[tid] = 0.f; s_cnt[tid] = 0.f; s_size[tid] = 0.f; s_first[tid] = NPT;
    if (tid < 8) s_scal[tid] = 0.f;
    __syncthreads();

    float ls0=0,ls1=0,ls2=0,ls3=0,ls4=0,ls5=0,ls6=0,ls7=0;
    long base = (long)e * NPT;
    int n0 = blockIdx.x * 4096;                 // gridDim.x == 64
    for (int p = 0; p < 16; ++p) {
        int n = n0 + p * 256 + tid;
        float bb = beta[base + n];
        int   s  = sid[base + n];
        bool  c  = cp[base + n] != 0;
        bool  valid = s >= 0;
        float pr = 1.f / (1.f + expf(-bb));
        float ax = fabsf(bb);
        float lg = log1pf(expf(-ax));
        float ce0 = fmaxf(bb, 0.f) + lg;        // softplus(b)
        float ce1 = fmaxf(-bb, 0.f) + lg;       // softplus(-b)
        if (c && valid) {
            float om = 1.f - pr;
            atomicAdd(&s_sumf[s], ALPHA_C * om * om * ce1);
            atomicAdd(&s_cnt[s], 1.f);
            atomicMin(&s_first[s], n);
        }
        if (valid) { atomicAdd(&s_size[s], 1.f); ls0 += 1.f; }
        if (c) { ls1 += 1.f; ls4 += fmaxf(THR + MARGIN_C - pr, 0.f); }
        else   { ls2 += 1.f; ls3 += ce0; ls5 += fmaxf(pr - (THR - MARGIN_C), 0.f); }
        if (s == -1) { ls6 += 1.f; ls7 += ce0; }
    }
    atomicAdd(&s_scal[0], ls0); atomicAdd(&s_scal[1], ls1);
    atomicAdd(&s_scal[2], ls2); atomicAdd(&s_scal[3], ls3);
    atomicAdd(&s_scal[4], ls4); atomicAdd(&s_scal[5], ls5);
    atomicAdd(&s_scal[6], ls6); atomicAdd(&s_scal[7], ls7);
    __syncthreads();

    float* ev = ws + (size_t)e * EV_WORDS;
    int*   evi = (int*)ev;
    if (s_cnt[tid] != 0.f) {
        atomicAdd(&ev[OFF_SUMF + tid], s_sumf[tid]);
        atomicAdd(&ev[OFF_CNT  + tid], s_cnt[tid]);
    }
    if (s_size[tid] != 0.f) atomicAdd(&ev[OFF_SIZE + tid], s_size[tid]);
    if (s_first[tid] < NPT) atomicMin(&evi[OFF_FIRST + tid], s_first[tid]);
    if (tid < 8) atomicAdd(&ev[OFF_SCAL + tid], s_scal[tid]);
}

// ---- ordered CP compaction (first M_MAX valid CPs in index order) + gather ----
__global__ __launch_bounds__(1024) void k_compact(const float* __restrict__ embed,
                                                  const int* __restrict__ sid,
                                                  const unsigned char* __restrict__ cp,
                                                  float* __restrict__ ws) {
    __shared__ int scan[1024];
    __shared__ int slist[MMAX];
    int tid = threadIdx.x;
    int e = blockIdx.x;
    long base = (long)e * NPT;
    int running = 0;
    for (int c0 = 0; c0 < NPT; c0 += 1024) {
        int n = c0 + tid;
        int flag = (sid[base + n] >= 0 && cp[base + n] != 0) ? 1 : 0;
        scan[tid] = flag;
        __syncthreads();
        for (int off = 1; off < 1024; off <<= 1) {
            int v = (tid >= off) ? scan[tid - off] : 0;
            __syncthreads();
            scan[tid] += v;
            __syncthreads();
        }
        int pos = running + scan[tid] - flag;
        if (flag && pos < MMAX) slist[pos] = n;
        running += scan[1023];
        __syncthreads();
    }
    int total = running;
    int capped = total < MMAX ? total : MMAX;
    float* ev = ws + (size_t)e * EV_WORDS;
    int*   evi = (int*)ev;
    if (tid == 0) { ev[OFF_SCAL + 10] = (float)total; evi[OFF_SCAL + 13] = capped; }
    __syncthreads();
    int m = tid;  // blockDim == MMAX
    if (m < capped) {
        int idx = slist[m];
        const float* er = embed + (base + idx) * DIM;
        float n2 = 0.f;
        #pragma unroll
        for (int d = 0; d < DIM; ++d) { float v = er[d]; ev[OFF_CPEMB + m*DIM + d] = v; n2 += v*v; }
        ev[OFF_N2 + m] = n2; ev[OFF_SELV + m] = 1.f;
    } else {
        #pragma unroll
        for (int d = 0; d < DIM; ++d) ev[OFF_CPEMB + m*DIM + d] = 0.f;
        ev[OFF_N2 + m] = 0.f; ev[OFF_SELV + m] = 0.f;
    }
}

// ---- attraction: d2 to per-instance anchor, LDS anchor table ----
__global__ __launch_bounds__(256) void k_attract(const float* __restrict__ embed,
                                                 const int* __restrict__ sid,
                                                 float* __restrict__ ws) {
    __shared__ float anc[KSEG * DIM];
    __shared__ float scale[KSEG];
    __shared__ float red[256];
    int tid = threadIdx.x;
    int e = blockIdx.y;
    float* ev = ws + (size_t)e * EV_WORDS;
    const int* evi = (const int*)ev;
    long base = (long)e * NPT;
    {
        int k = tid;
        int fi = evi[OFF_FIRST + k];
        float has = (ev[OFF_CNT + k] > 0.f) ? 1.f : 0.f;
        int src = fi < NPT ? fi : NPT - 1;
        const float* er = embed + (base + src) * DIM;
        #pragma unroll
        for (int d = 0; d < DIM; ++d) anc[k*DIM + d] = er[d];
        scale[k] = has / fmaxf(ev[OFF_SIZE + k], 1.f);
    }
    __syncthreads();
    float acc = 0.f;
    int n0 = blockIdx.x * 4096;
    for (int p = 0; p < 16; ++p) {
        int n = n0 + p * 256 + tid;
        int s = sid[base + n];
        if (s >= 0) {
            const float* er = embed + (base + n) * DIM;
            const float* ar = &anc[s * DIM];
            float d2 = 0.f;
            #pragma unroll
            for (int d = 0; d < DIM; ++d) { float df = er[d] - ar[d]; d2 += df * df; }
            acc += d2 * scale[s];
        }
    }
    red[tid] = acc; __syncthreads();
    for (int s2 = 128; s2 > 0; s2 >>= 1) { if (tid < s2) red[tid] += red[tid + s2]; __syncthreads(); }
    if (tid == 0) atomicAdd(&ev[OFF_SCAL + 8], red[0]);
}

// ---- repulsion: Gram via V_WMMA_F32_16X16X4_F32, pd2 = ni + nj - 2*G ----
__global__ __launch_bounds__(512) void k_repulse(float* __restrict__ ws) {
    __shared__ float Csh[MMAX * DIM];   // 64 KB: CP matrix, fp32
    __shared__ float n2sh[MMAX];
    __shared__ float svsh[MMAX];
    __shared__ float red[512];
    int tid = threadIdx.x;
    int e = blockIdx.y;
    float* ev = ws + (size_t)e * EV_WORDS;
    for (int i = tid; i < MMAX * DIM; i += 512) Csh[i] = ev[OFF_CPEMB + i];
    for (int i = tid; i < MMAX; i += 512) { n2sh[i] = ev[OFF_N2 + i]; svsh[i] = ev[OFF_SELV + i]; }
    __syncthreads();

    int lane = tid & 31, wv = tid >> 5;
    int l = lane & 15, h = lane >> 4;   // f32 16x4 A/B layout: lanes 0-15 hold K0/K1, 16-31 hold K2/K3
    float lsum = 0.f;
    for (int q = 0; q < 8; ++q) {
        int t = blockIdx.x * 128 + wv * 8 + q;   // 4096 tiles over gridDim.x==32 blocks
        int ti = t >> 6, tj = t & 63;
        const float* Ar = &Csh[(ti * 16 + l) * DIM];
        const float* Br = &Csh[(tj * 16 + l) * DIM];
        v8f acc = {0.f,0.f,0.f,0.f,0.f,0.f,0.f,0.f};
        #pragma unroll
        for (int kk = 0; kk < 4; ++kk) {
            v2f a = { Ar[kk*4 + h*2], Ar[kk*4 + h*2 + 1] };
            v2f b = { Br[kk*4 + h*2], Br[kk*4 + h*2 + 1] };
            acc = __builtin_amdgcn_wmma_f32_16x16x4_f32(false, a, false, b, (short)0, acc, false, false);
        }
        float n2j = n2sh[tj * 16 + l];
        float svj = svsh[tj * 16 + l];
        #pragma unroll
        for (int r = 0; r < 8; ++r) {
            int i = ti * 16 + r + h * 8;
            float pd2 = fmaxf(n2sh[i] + n2j - 2.f * acc[r], 0.f);
            lsum += svsh[i] * svj * expf(-pd2);
        }
    }
    red[tid] = lsum; __syncthreads();
    for (int s2 = 256; s2 > 0; s2 >>= 1) { if (tid < s2) red[tid] += red[tid + s2]; __syncthreads(); }
    if (tid == 0) atomicAdd(&ev[OFF_SCAL + 9], red[0]);
}

// ---- per-event finalize ----
__global__ __launch_bounds__(256) void k_finalize(float* __restrict__ ws) {
    __shared__ float r1[256], r2[256];
    int tid = threadIdx.x;
    int e = blockIdx.x;
    float* ev = ws + (size_t)e * EV_WORDS;
    float cnt = ev[OFF_CNT + tid], sz = ev[OFF_SIZE + tid], sf = ev[OFF_SUMF + tid];
    float has = cnt > 0.f ? 1.f : 0.f;
    float w = has * sz;
    r1[tid] = w * (sf / fmaxf(cnt, 1.f));
    r2[tid] = w;
    __syncthreads();
    for (int s = 128; s > 0; s >>= 1) { if (tid < s) { r1[tid] += r1[tid+s]; r2[tid] += r2[tid+s]; } __syncthreads(); }
    if (tid == 0) {
        float pos_bce = r1[0] / fmaxf(r2[0], 1.f);
        float n_valid = ev[OFF_SCAL+0], n_cp = ev[OFF_SCAL+1], n_ncp = ev[OFF_SCAL+2];
        float ce0ncp = ev[OFF_SCAL+3], posm = ev[OFF_SCAL+4], negm = ev[OFF_SCAL+5];
        float n_bg = ev[OFF_SCAL+6], bg = ev[OFF_SCAL+7];
        float attr = ev[OFF_SCAL+8], rep = ev[OFF_SCAL+9], mv = ev[OFF_SCAL+10];
        float neg_bce    = n_ncp > 0.f ? ce0ncp / fmaxf(n_ncp, 1.f) : 0.f;
        float pos_margin = posm / fmaxf(n_cp, 1.f);
        float neg_margin = n_ncp > 0.f ? negm / fmaxf(n_ncp, 1.f) : 0.f;
        float bg_bce     = n_bg > 0.f ? bg / fmaxf(n_bg, 1.f) : 0.f;
        float beta_loss = BETA_POS_W * pos_bce + BETA_NEG_SIG_W * neg_bce +
                          BETA_NEG_BG_W * bg_bce + MARGIN_W * (pos_margin + neg_margin);
        float repl = mv > 1.f ? REP_W * rep / fmaxf(mv * mv, 1.f) : 0.f;
        ev[OFF_SCAL + 11] = beta_loss + ATTR_W * attr + repl;
        ev[OFF_SCAL + 12] = (n_valid > 0.f && mv > 0.f) ? 1.f : 0.f;
    }
}

__global__ void k_combine(const float* __restrict__ ws, float* __restrict__ out) {
    if (threadIdx.x == 0) {
        float tot = 0.f, c = 0.f;
        for (int e = 0; e < BEV; ++e) {
            const float* ev = ws + (size_t)e * EV_WORDS;
            if (ev[OFF_SCAL + 12] > 0.f) { tot += ev[OFF_SCAL + 11]; c += 1.f; }
        }
        out[0] = c > 0.f ? tot / fmaxf(c, 1.f) : 0.f;
    }
}

extern "C" void kernel_launch(void* const* d_in, const int* in_sizes, int n_in,
                              void* d_out, int out_size, void* d_ws, size_t ws_size,
                              hipStream_t stream) {
    const float*         beta  = (const float*)d_in[0];
    const float*         embed = (const float*)d_in[1];
    const int*           sid   = (const int*)d_in[2];
    const unsigned char* cp    = (const unsigned char*)d_in[3];
    float* out = (float*)d_out;
    float* ws  = (float*)d_ws;
    (void)in_sizes; (void)n_in; (void)out_size; (void)ws_size;

    int totalWords = BEV * EV_WORDS;
    k_init    <<<(totalWords + 255) / 256, 256, 0, stream>>>(ws);
    k_stats   <<<dim3(64, BEV), 256, 0, stream>>>(beta, sid, cp, ws);
    k_compact <<<BEV, 1024, 0, stream>>>(embed, sid, cp, ws);
    k_attract <<<dim3(64, BEV), 256, 0, stream>>>(embed, sid, ws);
    k_repulse <<<dim3(32, BEV), 512, 0, stream>>>(ws);
    k_finalize<<<BEV, 256, 0, stream>>>(ws);
    k_combine <<<1, 32, 0, stream>>>(ws, out);
}